// reaction_diffusion_nature_fml_90872918049154
// MI455X (gfx1250) — compile-verified
//
#include <hip/hip_runtime.h>
#include <hip/hip_bf16.h>
#include <math.h>

#define NN    207   // nodes
#define NE    1722  // edges
#define BB    64    // batch
#define PAD   224   // 14 * 16 : K and N padded for 16x16x4 WMMA tiling

typedef __attribute__((ext_vector_type(2))) float v2f;
typedef __attribute__((ext_vector_type(8))) float v8f;

// ---------------------------------------------------------------------------
// Kernel 0: zero the two padded transposed-Laplacian buffers in workspace.
// ---------------------------------------------------------------------------
__global__ void rd_zero_ws(float* __restrict__ w, int n) {
    int i = blockIdx.x * blockDim.x + threadIdx.x;
    if (i < n) w[i] = 0.0f;
}

// ---------------------------------------------------------------------------
// Kernel 1: scatter-build B = L^T for both Laplacians.
//   reaction : W[i,j] += wr ; Lr = diag(colsum) - W ; Br[k,n] = Lr[n,k]
//     => per edge: Br[j*PAD + i] -= wr ; Br[j*PAD + j] += wr
//   diffusion: W[j,i] += wd ; Ld = diag(colsum) - W ; Bd[k,n] = Ld[n,k]
//     => per edge: Bd[i*PAD + j] -= wd ; Bd[i*PAD + i] += wd
// ---------------------------------------------------------------------------
__global__ void rd_build_laplacians(const float* __restrict__ wr,
                                    const float* __restrict__ wd,
                                    const int*   __restrict__ ei,
                                    const int*   __restrict__ ej,
                                    float* __restrict__ Br,
                                    float* __restrict__ Bd) {
    int k = blockIdx.x * blockDim.x + threadIdx.x;
    if (k >= NE) return;
    int i = ei[k], j = ej[k];
    float r = wr[k], d = wd[k];
    atomicAdd(&Br[j * PAD + i], -r);
    atomicAdd(&Br[j * PAD + j],  r);
    atomicAdd(&Bd[i * PAD + j], -d);
    atomicAdd(&Bd[i * PAD + i],  d);
}

// ---------------------------------------------------------------------------
// Kernel 2: fused dual WMMA matmul + epilogue.
// One wave32 per 16x16 output tile; grid = (4 M-tiles, 14 N-tiles).
// A fragment loaded once per K-step, fed to two v_wmma_f32_16x16x4_f32
// (reaction + diffusion accumulators).
// ---------------------------------------------------------------------------
__global__ __launch_bounds__(32)
void rd_wmma(const float* __restrict__ input,
             const float* __restrict__ Br,
             const float* __restrict__ Bd,
             const float* __restrict__ bias_r,
             const float* __restrict__ bias_d,
             float* __restrict__ out) {
    __shared__ float sA[16 * PAD];           // 14 KB: padded 16-row A slice

    const int mt   = blockIdx.x;             // 0..3   (batch tile)
    const int nt   = blockIdx.y;             // 0..13  (node tile)
    const int lane = threadIdx.x;            // 0..31
    const int half = lane >> 4;              // 0 or 1 (lane half)
    const int l16  = lane & 15;

    // Stage zero-padded A tile: rows mt*16..+15, cols 0..223 (0 for c>=207).
    // Uniform trip count across lanes -> EXEC stays all-1s.
    for (int idx = lane; idx < 16 * PAD; idx += 32) {
        int r = idx / PAD, c = idx % PAD;
        sA[idx] = (c < NN) ? input[(mt * 16 + r) * NN + c] : 0.0f;
    }
    __syncthreads();

    v8f cr = {};                             // reaction accumulator (16x16 f32)
    v8f cd = {};                             // diffusion accumulator
    const int ncol = nt * 16 + l16;          // this lane's N column (0..223)

    // K loop over padded depth, 4 per WMMA step, branch-free (zero padding
    // in both sA and the workspace B matrices makes padded steps exact).
    #pragma unroll 4
    for (int kb = 0; kb < PAD; kb += 4) {
        const int ka = kb + 2 * half;        // ISA A layout: halves hold K {0,1}/{2,3}
        v2f a;
        a.x = sA[l16 * PAD + ka];
        a.y = sA[l16 * PAD + ka + 1];
        v2f fr;                              // B layout: row K striped across lanes
        fr.x = Br[ka * PAD + ncol];
        fr.y = Br[(ka + 1) * PAD + ncol];
        v2f fd;
        fd.x = Bd[ka * PAD + ncol];
        fd.y = Bd[(ka + 1) * PAD + ncol];
        cr = __builtin_amdgcn_wmma_f32_16x16x4_f32(false, a, false, fr,
                                                   (short)0, cr, false, false);
        cd = __builtin_amdgcn_wmma_f32_16x16x4_f32(false, a, false, fd,
                                                   (short)0, cd, false, false);
    }

    // Epilogue: out = tanh(reaction + br) + diffusion + bd + input.
    // C/D layout: VGPR v, lane half h -> row M = v + 8*h, col N = lane&15.
    if (ncol < NN) {
        const float brv = bias_r[ncol];
        const float bdv = bias_d[ncol];
        #pragma unroll
        for (int v = 0; v < 8; ++v) {
            int row = v + 8 * half;
            int b   = mt * 16 + row;
            float x = sA[row * PAD + ncol];          // input[b, ncol]
            out[b * NN + ncol] = tanhf(cr[v] + brv) + cd[v] + bdv + x;
        }
    }
}

// ---------------------------------------------------------------------------
extern "C" void kernel_launch(void* const* d_in, const int* in_sizes, int n_in,
                              void* d_out, int out_size, void* d_ws, size_t ws_size,
                              hipStream_t stream) {
    const float* input   = (const float*)d_in[0];  // [64, 207]
    const float* wr      = (const float*)d_in[1];  // [1, 1722]
    const float* wd      = (const float*)d_in[2];  // [1, 1722]
    const float* bias_r  = (const float*)d_in[3];  // [1, 207]
    const float* bias_d  = (const float*)d_in[4];  // [1, 207]
    const int*   ei      = (const int*)d_in[5];    // [1722]
    const int*   ej      = (const int*)d_in[6];    // [1722]
    float*       out     = (float*)d_out;          // [64, 207]

    float* Br = (float*)d_ws;                      // [PAD, PAD] = Lr^T padded
    float* Bd = Br + PAD * PAD;                    // [PAD, PAD] = Ld^T padded

    const int ztot = 2 * PAD * PAD;
    rd_zero_ws<<<(ztot + 255) / 256, 256, 0, stream>>>(Br, ztot);
    rd_build_laplacians<<<(NE + 255) / 256, 256, 0, stream>>>(wr, wd, ei, ej, Br, Bd);

    dim3 grid(BB / 16, PAD / 16);                  // (4, 14) tiles
    rd_wmma<<<grid, 32, 0, stream>>>(input, Br, Bd, bias_r, bias_d, out);
}